// IndRNN_3813930959120
// MI455X (gfx1250) — compile-verified
//
#include <hip/hip_runtime.h>

// Problem constants (match reference)
static constexpr int B_ = 64;
static constexpr int T_ = 2048;
static constexpr int I_ = 256;
static constexpr int H_ = 512;
static constexpr long long BT_ = (long long)B_ * T_;   // 131072 rows

typedef __attribute__((ext_vector_type(2))) float v2f;
typedef __attribute__((ext_vector_type(4))) float v4f;
typedef __attribute__((ext_vector_type(8))) float v8f;
typedef __attribute__((ext_vector_type(4))) int   v4i;

#define KC   64    // K-chunk staged in LDS
#define WROW 68    // padded LDS row stride (floats): bank-conflict-free b64 reads

#if __has_builtin(__builtin_amdgcn_global_load_async_to_lds_b128)
#define HAVE_ASYNC_LDS 1
#else
#define HAVE_ASYNC_LDS 0
#endif

#if HAVE_ASYNC_LDS
typedef __attribute__((address_space(1))) v4i gas_v4i;  // global (AS1)
typedef __attribute__((address_space(3))) v4i las_v4i;  // LDS (AS3)
__device__ __forceinline__ void async_copy_b128(const float* g, float* l) {
  __builtin_amdgcn_global_load_async_to_lds_b128(
      (gas_v4i*)g, (las_v4i*)l, 0, 0);
}
#endif

__device__ __forceinline__ void wait_async_zero() {
#if __has_builtin(__builtin_amdgcn_s_wait_asynccnt)
  __builtin_amdgcn_s_wait_asynccnt(0);
#else
  asm volatile("s_wait_asynccnt 0x0" ::: "memory");
#endif
}

// C = A(MxK, row-major) * W^T (W is NxK row-major) + bias(N), all f32 exact.
// Block = 256 threads = 8 waves. Wave -> 16x64 C strip, block -> 128x64.
// W tile (64 x KC) is staged in LDS (async DMA, double buffered); A stays in
// registers (streamed, per-wave private). grid.x = N/64, grid.y = M/128.
__global__ __launch_bounds__(256) void gemm_bias_wmma_f32(
    const float* __restrict__ A, const float* __restrict__ W,
    const float* __restrict__ bias, float* __restrict__ C,
    int N, int K) {
  __shared__ float sw[2][64 * WROW];

  const int tid  = threadIdx.x;
  const int lane = tid & 31;
  const int wave = tid >> 5;
  const int l15  = lane & 15;
  const int half = lane >> 4;                  // 0: lanes 0-15, 1: lanes 16-31
  const long long m0 = ((long long)blockIdx.y * 8 + wave) * 16;
  const int n0 = blockIdx.x * 64;

  // A fragment row base: a.x = A[m0+l15, k+2*half], a.y = A[m0+l15, k+2*half+1]
  const float* arow = A + (m0 + l15) * (long long)K + 2 * half;

  // Cooperative W-tile copy: 64 rows x KC floats = 16KB per chunk.
  // thread -> row = tid>>2, segments s = (tid&3) + 4*i (16B each).
  const int crow  = tid >> 2;                  // 0..63
  const int cseg0 = tid & 3;                   // 0..3

  auto stage_w = [&](int buf, int kc0) {
#if HAVE_ASYNC_LDS
#pragma unroll
    for (int i = 0; i < 4; ++i) {
      const int seg = cseg0 + 4 * i;           // 0..15
      const float* g = W + (long long)(n0 + crow) * K + kc0 + seg * 4;
      float* l = &sw[buf][crow * WROW + seg * 4];
      async_copy_b128(g, l);
    }
#else
#pragma unroll
    for (int i = 0; i < 4; ++i) {
      const int seg = cseg0 + 4 * i;
      const v4f g = *(const v4f*)(W + (long long)(n0 + crow) * K + kc0 + seg * 4);
      *(v4f*)&sw[buf][crow * WROW + seg * 4] = g;
    }
#endif
  };

  v8f acc0 = {}, acc1 = {}, acc2 = {}, acc3 = {};

  stage_w(0, 0);
#if HAVE_ASYNC_LDS
  wait_async_zero();
#endif
  __syncthreads();

  const int nchunks = K / KC;
  for (int c = 0; c < nchunks; ++c) {
    const int kc0 = c * KC;
    // Kick off DMA for the next chunk; it overlaps the 64 WMMAs below.
    if (c + 1 < nchunks) stage_w((c + 1) & 1, kc0 + KC);

    // Per-wave A chunk: 16 independent b64 loads issued as one batch.
    v2f a[16];
#pragma unroll
    for (int i = 0; i < 16; ++i) a[i] = *(const v2f*)(arow + kc0 + 4 * i);

    const float* wb = &sw[c & 1][0];
    // B fragment j: b.x = W[n0+16j+l15, k+2*half] == B[k,n]; rotated prefetch.
    v2f b[4], bn[4];
#pragma unroll
    for (int j = 0; j < 4; ++j)
      b[j] = *(const v2f*)(wb + (16 * j + l15) * WROW + 2 * half);

#pragma unroll
    for (int s = 0; s < 16; ++s) {
      if (s + 1 < 16) {
#pragma unroll
        for (int j = 0; j < 4; ++j)
          bn[j] = *(const v2f*)(wb + (16 * j + l15) * WROW + 2 * half + 4 * (s + 1));
      }
      acc0 = __builtin_amdgcn_wmma_f32_16x16x4_f32(false, a[s], false, b[0], (short)0, acc0, false, false);
      acc1 = __builtin_amdgcn_wmma_f32_16x16x4_f32(false, a[s], false, b[1], (short)0, acc1, false, false);
      acc2 = __builtin_amdgcn_wmma_f32_16x16x4_f32(false, a[s], false, b[2], (short)0, acc2, false, false);
      acc3 = __builtin_amdgcn_wmma_f32_16x16x4_f32(false, a[s], false, b[3], (short)0, acc3, false, false);
#pragma unroll
      for (int j = 0; j < 4; ++j) b[j] = bn[j];
    }

#if HAVE_ASYNC_LDS
    wait_async_zero();   // next chunk's DMA landed
#endif
    __syncthreads();     // all waves done reading buf (c&1) before reuse
  }

  // C layout: acc[j][r] -> row m0 + r + 8*half, col n0 + 16*j + l15
  const long long rbase = m0 + 8 * half;
  v8f accs[4] = {acc0, acc1, acc2, acc3};
#pragma unroll
  for (int j = 0; j < 4; ++j) {
    const int n = n0 + 16 * j + l15;
    const float bv = bias[n];
    float* cptr = C + rbase * N + n;
#pragma unroll
    for (int r = 0; r < 8; ++r) {
      cptr[(long long)r * N] = accs[j][r] + bv;
    }
  }
}

// h_t = relu(pre[:,t,:] + w_hh * h_{t-1}); one thread per (b,h).
// pre loads are independent of the recurrence -> 16 loads in flight while the
// 2-op dependent chain retires; prefetch one unroll-block ahead.
__global__ __launch_bounds__(256) void indrnn_scan(
    const float* __restrict__ pre, const float* __restrict__ w_hh,
    float* __restrict__ out, int T, int H) {
  const int idx = blockIdx.x * blockDim.x + threadIdx.x;  // over B*H
  const int b = idx / H;
  const int h = idx - b * H;
  const float w = w_hh[h];
  const float* p = pre + (long long)b * T * H + h;
  float*       o = out + (long long)b * T * H + h;

  float s = 0.0f;
  constexpr int U = 16;
  for (int t = 0; t < T; t += U) {
    if (t + U < T) __builtin_prefetch(&p[(long long)(t + U) * H], 0, 1);
    float v[U];
#pragma unroll
    for (int u = 0; u < U; ++u) v[u] = p[(long long)(t + u) * H];
#pragma unroll
    for (int u = 0; u < U; ++u) {
      s = fmaxf(fmaf(w, s, v[u]), 0.0f);
      o[(long long)(t + u) * H] = s;
    }
  }
}

extern "C" void kernel_launch(void* const* d_in, const int* in_sizes, int n_in,
                              void* d_out, int out_size, void* d_ws, size_t ws_size,
                              hipStream_t stream) {
  const float* x     = (const float*)d_in[0];  // (B,T,I)
  const float* w_ih0 = (const float*)d_in[1];  // (H,I)
  const float* w_hh0 = (const float*)d_in[2];  // (H)
  const float* b_ih0 = (const float*)d_in[3];  // (H)
  const float* w_ih1 = (const float*)d_in[4];  // (H,H)
  const float* w_hh1 = (const float*)d_in[5];  // (H)
  const float* b_ih1 = (const float*)d_in[6];  // (H)
  float* out = (float*)d_out;                  // (B,T,H)

  float* pre  = (float*)d_ws;                  // BT*H floats (reused for layer 1)
  float* out0 = pre + BT_ * H_;                // BT*H floats

  dim3 gblk(256);
  dim3 ggrid(H_ / 64, (unsigned)(BT_ / 128));
  dim3 sblk(256);
  dim3 sgrid((unsigned)((B_ * H_) / 256));

  // Layer 0: pre0 = x @ w_ih0^T + b_ih0  (K = I = 256)
  gemm_bias_wmma_f32<<<ggrid, gblk, 0, stream>>>(x, w_ih0, b_ih0, pre, H_, I_);
  // Layer 0 scan -> out0
  indrnn_scan<<<sgrid, sblk, 0, stream>>>(pre, w_hh0, out0, T_, H_);
  // Layer 1: pre1 = out0 @ w_ih1^T + b_ih1  (K = H = 512), reuse `pre`
  gemm_bias_wmma_f32<<<ggrid, gblk, 0, stream>>>(out0, w_ih1, b_ih1, pre, H_, H_);
  // Layer 1 scan -> final output
  indrnn_scan<<<sgrid, sblk, 0, stream>>>(pre, w_hh1, out, T_, H_);
}